// GwcVolumeCostProcessor_78451872629273
// MI455X (gfx1250) — compile-verified
//
#include <hip/hip_runtime.h>

typedef __attribute__((ext_vector_type(2))) float v2f;
typedef __attribute__((ext_vector_type(4))) float v4f;
typedef __attribute__((ext_vector_type(8))) float v8f;

#define HH      128
#define WW      240
#define GROUPS  40
#define CPG     8
#define DD      48

#define LDS_PITCH 20            // floats per row: 16B-aligned rows, bank-friendly
#define LDS_ROWS  80            // rows map d in [-16, 64) -> row d+16

// One wave (32 lanes) per (group g, row h, 16-wide w tile).
// gwc[g,d,h,w] = 1/8 * sum_c ref[g*8+c,h,w] * tgt[g*8+c,h,w-d], zero for w<d.
// Banded matmul: A = ref tile [16w x 8ch] (pre-scaled by 1/8),
// B = tgt tile [8ch x 16w'], D[M][N] lies on diagonal d = off + M - N.
// Two chained V_WMMA_F32_16X16X4_F32 per offset tile, 4 offsets cover
// d in [0,48) exactly once; out-of-band diagonals fall into LDS pad rows.
__global__ __launch_bounds__(32) void gwc_wmma_kernel(
    const float* __restrict__ ref, const float* __restrict__ tgt,
    float* __restrict__ out)
{
  const int wt   = blockIdx.x;      // 0..14  (240/16 w tiles)
  const int h    = blockIdx.y;      // 0..127
  const int g    = blockIdx.z;      // 0..39
  const int wb   = wt << 4;
  const int lane = threadIdx.x;     // 0..31
  const int hf   = lane >> 4;       // lane half
  const int lm   = lane & 15;

  __shared__ __align__(16) float lds[LDS_ROWS * LDS_PITCH];

  const int planeHW = HH * WW;
  const int base_gh = (g * CPG) * planeHW + h * WW;

  // A matrix (ref): M = lm (w index), K = channel, pre-scaled by 1/8 (mean).
  // f32 16x4 A layout: VGPR0 = K0(lanes0-15)/K2(lanes16-31); VGPR1 = K1/K3.
  const int wm = wb + lm;
  const int ka = 2 * hf;            // this lane's K channel base
  v2f aLo, aHi;
  aLo.x = ref[base_gh + (ka + 0) * planeHW + wm] * 0.125f;
  aLo.y = ref[base_gh + (ka + 1) * planeHW + wm] * 0.125f;
  aHi.x = ref[base_gh + (ka + 4) * planeHW + wm] * 0.125f;
  aHi.y = ref[base_gh + (ka + 5) * planeHW + wm] * 0.125f;

  #pragma unroll
  for (int o = 0; o < 64; o += 16) {
    // B matrix (tgt): N = lm, w' = wb - o + lm. Clamp address (always legal),
    // select value -> unconditional loads, no divergent branches.
    const int  wp  = wb - o + lm;
    const bool ok  = (wp >= 0);
    const int  wpc = ok ? wp : 0;
    const float t0 = tgt[base_gh + (ka + 0) * planeHW + wpc];
    const float t1 = tgt[base_gh + (ka + 1) * planeHW + wpc];
    const float t2 = tgt[base_gh + (ka + 4) * planeHW + wpc];
    const float t3 = tgt[base_gh + (ka + 5) * planeHW + wpc];
    v2f bLo, bHi;
    bLo.x = ok ? t0 : 0.0f;
    bLo.y = ok ? t1 : 0.0f;
    bHi.x = ok ? t2 : 0.0f;
    bHi.y = ok ? t3 : 0.0f;

    v8f c = {0.f, 0.f, 0.f, 0.f, 0.f, 0.f, 0.f, 0.f};
    c = __builtin_amdgcn_wmma_f32_16x16x4_f32(false, aLo, false, bLo,
                                              (short)0, c, false, false);
    c = __builtin_amdgcn_wmma_f32_16x16x4_f32(false, aHi, false, bHi,
                                              (short)0, c, false, false);

    // C/D layout: lane half selects M+8; VGPR r = M row. d = o + M - N.
    // Unconditional scatter: d in (o-16, o+16) always lands in rows [0,80).
    #pragma unroll
    for (int r = 0; r < 8; ++r) {
      const int M = r + 8 * hf;
      const int d = o + M - lm;                  // may be <0 or >=48: pad rows
      lds[(d + 16) * LDS_PITCH + M] = c[r];
    }
  }

  __syncthreads();

  // Coalesced writeback: 48 rows x 16 floats, v4f per lane, streaming (NT).
  #pragma unroll
  for (int i = 0; i < 6; ++i) {
    const int flat = i * 32 + lane;              // 0..191 = d*4 + q
    const int d = flat >> 2;
    const int q = flat & 3;
    const v4f v = *(const v4f*)&lds[(d + 16) * LDS_PITCH + q * 4];
    v4f* dst = (v4f*)(out + ((size_t)(g * DD + d) * HH + h) * WW + wb + q * 4);
    __builtin_nontemporal_store(v, dst);
  }
}

// Concat volume: out channels 40..51 = masked ref_concat, 52..63 = shifted tgt_concat.
__global__ __launch_bounds__(256) void concat_kernel(
    const float* __restrict__ refc, const float* __restrict__ tgtc,
    float* __restrict__ out)
{
  const int d  = blockIdx.x;   // 0..47
  const int h  = blockIdx.y;   // 0..127
  const int c2 = blockIdx.z;   // 0..23
  const int w  = threadIdx.x;
  if (w >= WW) return;

  float val;
  if (c2 < 12) {
    val = (w >= d) ? refc[(c2 * HH + h) * WW + w] : 0.0f;
  } else {
    val = (w >= d) ? tgtc[((c2 - 12) * HH + h) * WW + (w - d)] : 0.0f;
  }
  __builtin_nontemporal_store(
      val, &out[(((size_t)(GROUPS + c2) * DD + d) * HH + h) * WW + w]);
}

extern "C" void kernel_launch(void* const* d_in, const int* in_sizes, int n_in,
                              void* d_out, int out_size, void* d_ws, size_t ws_size,
                              hipStream_t stream) {
  const float* ref_gwc    = (const float*)d_in[0];
  const float* tgt_gwc    = (const float*)d_in[1];
  const float* ref_concat = (const float*)d_in[2];
  const float* tgt_concat = (const float*)d_in[3];
  float* out = (float*)d_out;

  dim3 gridA(WW / 16, HH, GROUPS);     // 15 x 128 x 40 single-wave blocks
  gwc_wmma_kernel<<<gridA, 32, 0, stream>>>(ref_gwc, tgt_gwc, out);

  dim3 gridB(DD, HH, 24);
  concat_kernel<<<gridB, 256, 0, stream>>>(ref_concat, tgt_concat, out);
}